// WaveletPatcher_63264868270249
// MI455X (gfx1250) — compile-verified
//
#include <hip/hip_runtime.h>

namespace {
constexpr int T_ = 4096;
constexpr int N_ = 32;
constexpr int B_ = 64;
constexpr int P_ = 511;
constexpr int PC_ = 16;       // patches per chunk
constexpr int NCHUNK = 32;    // ceil(P_/PC_)
constexpr int LT_ = 192;      // LDS tile rows (time), covers 136 out-t + halos
constexpr int STR_ = 36;      // padded floats per row (16B-aligned, bank-friendly)
constexpr int OFF_ = 24;      // left halo (need 21 for 3-level db4 left reach)

// Reversed db4 decomposition filters: coeff(t) = sum_k F[k] * prev[t + d*(k-3)]
constexpr float FLO[8] = { 0.23037781330885523f, 0.7148465705525415f,
  0.6308807679295904f, -0.02798376941698385f, -0.18703481171888114f,
  0.030841381835986965f, 0.032883011666982945f, -0.010597401784997278f };
constexpr float FHI[8] = { -0.010597401784997278f, -0.032883011666982945f,
  0.030841381835986965f, 0.18703481171888114f, -0.02798376941698385f,
  -0.6308807679295904f, 0.7148465705525415f, -0.23037781330885523f };
}

typedef int   v4i_t __attribute__((ext_vector_type(4)));
typedef float v4f_t __attribute__((ext_vector_type(4)));
#define AS1_ __attribute__((address_space(1)))
#define AS3_ __attribute__((address_space(3)))

#if __has_builtin(__builtin_amdgcn_global_load_async_to_lds_b128)
#define USE_ASYNC_LDS 1
#else
#define USE_ASYNC_LDS 0
#endif

// One lowpass output row element: dst[l] = sum_k FLO[k] * src[l + D*(k-3)]
template <int D, bool CLAMP>
__device__ __forceinline__ void lowpass_row(const float* __restrict__ src,
                                            float* __restrict__ dst,
                                            int l, int n) {
  float acc = 0.f;
  #pragma unroll
  for (int k = 0; k < 8; ++k) {
    int lr = l + D * (k - 3);
    if (CLAMP) lr = lr < 0 ? 0 : (lr > LT_ - 1 ? LT_ - 1 : lr);  // edges unused
    acc = __builtin_fmaf(FLO[k], src[lr * STR_ + n], acc);
  }
  dst[l * STR_ + n] = acc;
}

__global__ __launch_bounds__(256) void swt_patch(const float* __restrict__ x,
                                                 float* __restrict__ out) {
  __shared__ float Xs[LT_ * STR_];
  __shared__ float A1[LT_ * STR_];
  __shared__ float A2[LT_ * STR_];

  const int tid = threadIdx.x;
  const int b = blockIdx.x >> 5;       // 32 chunks per batch
  const int chunk = blockIdx.x & 31;
  const int p0 = chunk * PC_;
  const int s = p0 * 8;                // first output timestep of this chunk
  const float* xb = x + (size_t)b * (T_ * N_);

  // ---- Phase 0: async-stage x tile (periodic wrap), 128B-contiguous rows ----
  #pragma unroll
  for (int i = 0; i < (LT_ * N_ / 4) / 256; ++i) {   // 6 iters, b128 per lane
    int idx = tid + 256 * i;
    int l  = idx >> 3;            // tile row
    int n0 = (idx & 7) << 2;      // channel group of 4
    int tg = (s - OFF_ + l + T_) & (T_ - 1);
    const float* g = xb + tg * N_ + n0;
    float* lp = Xs + l * STR_ + n0;
#if USE_ASYNC_LDS
    __builtin_amdgcn_global_load_async_to_lds_b128(
        (AS1_ v4i_t*)(AS1_ void*)g,
        (AS3_ v4i_t*)(AS3_ void*)lp, 0, 0);
#else
    *(float4*)lp = *(const float4*)g;
#endif
  }
#if USE_ASYNC_LDS
#if __has_builtin(__builtin_amdgcn_s_wait_asynccnt)
  __builtin_amdgcn_s_wait_asynccnt(0);
#else
  asm volatile("s_wait_asynccnt 0x0" ::: "memory");
#endif
#endif
  __syncthreads();

  const int n  = tid & 31;   // lane = channel in recurrence passes
  const int lw = tid >> 5;

  // ---- Phase 1: level-1 lowpass A1 (dilation 1); clamp only on edge groups ----
  lowpass_row<1, true>(Xs, A1, lw, n);                 // rows 0..7
  #pragma unroll 4
  for (int i = 1; i < LT_ / 8 - 1; ++i)
    lowpass_row<1, false>(Xs, A1, lw + 8 * i, n);      // rows 8..183 (safe)
  lowpass_row<1, true>(Xs, A1, lw + (LT_ - 8), n);     // rows 184..191
  __syncthreads();

  // ---- Phase 2: level-2 lowpass A2 (dilation 2) ----
  lowpass_row<2, true>(A1, A2, lw, n);
  #pragma unroll 4
  for (int i = 1; i < LT_ / 8 - 1; ++i)
    lowpass_row<2, false>(A1, A2, lw + 8 * i, n);
  lowpass_row<2, true>(A1, A2, lw + (LT_ - 8), n);
  __syncthreads();

  // ---- Phase 3: evaluate patch coefficients, 4 outputs/thread, b128 NT stores ----
  // out element e: j = e&15, c = (e>>4)&3, n = (e>>6)&31, p = e>>11
  // c: 0=cA3(A2,lo,d4) 1=cD3(A2,hi,d4) 2=cD2(A1,hi,d2) 3=cD1(X,hi,d1)
  const int j0 = (tid & 3) << 2;         // j group of 4, constant per thread
  const int c  = (tid >> 2) & 3;         // constant per thread
  float f[8];
  #pragma unroll
  for (int k = 0; k < 8; ++k) f[k] = (c == 0) ? FLO[k] : FHI[k];
  const float* plane = (c < 2) ? A2 : ((c == 2) ? A1 : Xs);
  const int d = (c < 2) ? 4 : ((c == 2) ? 2 : 1);
  const int pcnt = (P_ - p0) < PC_ ? (P_ - p0) : PC_;
  const int eMax = pcnt * 2048;
  float* ob = out + (size_t)(b * P_ + p0) * 2048;   // chunk output is contiguous
  for (int i = 0; i < (PC_ * 2048) / 1024; ++i) {   // 32 iters
    int e0 = (tid << 2) + 1024 * i;                 // wave covers 512B contiguous
    if (e0 >= eMax) break;                          // eMax multiple of 2048
    int en = (e0 >> 6) & 31;
    int pl = e0 >> 11;
    int l0 = 8 * pl + j0 + OFF_;
    const float* pp = plane + en + (l0 - 3 * d) * STR_;
    v4f_t acc = {0.f, 0.f, 0.f, 0.f};
    #pragma unroll
    for (int k = 0; k < 8; ++k) {                   // identical addrs CSE for d<4
      const float* q = pp + k * d * STR_;
      acc.x = __builtin_fmaf(f[k], q[0],         acc.x);
      acc.y = __builtin_fmaf(f[k], q[STR_],      acc.y);
      acc.z = __builtin_fmaf(f[k], q[2 * STR_],  acc.z);
      acc.w = __builtin_fmaf(f[k], q[3 * STR_],  acc.w);
    }
    __builtin_nontemporal_store(acc, (v4f_t*)(ob + e0));
  }
}

extern "C" void kernel_launch(void* const* d_in, const int* in_sizes, int n_in,
                              void* d_out, int out_size, void* d_ws, size_t ws_size,
                              hipStream_t stream) {
  const float* x = (const float*)d_in[0];
  float* out = (float*)d_out;
  dim3 grid(B_ * NCHUNK);   // 2048 blocks
  dim3 block(256);          // 8 wave32
  hipLaunchKernelGGL(swt_patch, grid, block, 0, stream, x, out);
}